// AFT_FULL_64312840290603
// MI455X (gfx1250) — compile-verified
//
#include <hip/hip_runtime.h>
#include <hip/hip_bf16.h>

// ---------------------------------------------------------------------------
// AFT-full (B=32, N=1024, D=512) for MI455X / gfx1250, fp32 end-to-end.
//
// Bandwidth-bound (>=150MB HBM traffic @23.3TB/s ~ 15us floor vs ~1us of
// f32-WMMA math), so all GEMMs use V_WMMA_F32_16X16X4_F32 to stay faithful
// to the fp32 reference instead of downconverting.
//
// The dominant einsum kernel (aft_attn) uses the Tensor Data Mover:
// 32x64 f32 tiles of ek/ekv are DMA'd global->LDS via TENSOR_LOAD_TO_LDS
// with a double-buffered TENSORcnt pipeline, so compute waves only issue
// WMMAs + LDS fragment reads while the TDM streams the next chunk.
//
// Workspace layout (floats):
//   [0 .. 16M)        k   -> overwritten in place with ek = exp(k - kmax)
//   [16M .. 32M)      v   -> overwritten in place with ekv = ek * v
//   [32M .. 33M)      ew  = exp(pos_bias - rowmax)            (N*N)
//   [33M .. +16K)     kmax[b*D+d]
// total ~ 138.5 MB.  q is staged in d_out and consumed/overwritten by the
// fused attention epilogue.
// ---------------------------------------------------------------------------

typedef __attribute__((ext_vector_type(2))) float v2f;
typedef __attribute__((ext_vector_type(4))) float v4f;
typedef __attribute__((ext_vector_type(8))) float v8f;
typedef __attribute__((ext_vector_type(4))) unsigned int u32x4;
typedef __attribute__((ext_vector_type(8))) int i32x8;
typedef __attribute__((ext_vector_type(4))) int i32x4;

#define B_  32
#define N_  1024
#define D_  512

// ---------------------------------------------------------------------------
// TDM helper: async-load a 32(row) x 64(col) f32 tile from a row-major
// tensor with row stride 512 elements into LDS at lds_byte_off.
// D# built per CDNA5 ISA 8.3/8.4:
//   group0: count=1 | lds_addr | global_addr[56:0] | type=2
//   group1: data_size=4B, tensor_dim0=512, tensor_dim1=1024,
//           tile_dim0=64, tile_dim1=32, tensor_dim0_stride=512
//   groups 2/3: zero (2D tensor)
// ---------------------------------------------------------------------------
__device__ __forceinline__ void tdm_load_tile_32x64_f32(
    unsigned lds_byte_off, const float* __restrict__ gptr)
{
    const unsigned long long ga = (unsigned long long)(size_t)gptr;
    u32x4 g0;
    g0[0] = 1u;                                            // count=1 (user D#)
    g0[1] = lds_byte_off;                                  // lds_addr
    g0[2] = (unsigned)(ga & 0xFFFFFFFFu);                  // global_addr lo
    g0[3] = (unsigned)((ga >> 32) & 0x01FFFFFFu)           // global_addr hi
          | (2u << 30);                                    // type=2 (image)
    i32x8 g1;
    g1[0] = 0x20000;        // [17:16] data_size=2 -> 4 bytes
    g1[1] = 0x02000000;     // [63:48] tensor_dim0 = 512
    g1[2] = 0x04000000;     // [95:80] tensor_dim1 = 1024
    g1[3] = 64 << 16;       // [127:112] tile_dim0 = 64
    g1[4] = 32;             // [143:128] tile_dim1 = 32 (tile_dim2 = 0)
    g1[5] = 512;            // [191:160] tensor_dim0_stride = 512
    g1[6] = 0;              // dim0_stride hi / dim1_stride lo
    g1[7] = 0;              // dim1_stride hi
    const i32x4 z4 = {0, 0, 0, 0};
#if __clang_major__ >= 23
    const i32x8 z8 = {0, 0, 0, 0, 0, 0, 0, 0};
    __builtin_amdgcn_tensor_load_to_lds(g0, g1, z4, z4, z8, 0);
#else
    __builtin_amdgcn_tensor_load_to_lds(g0, g1, z4, z4, 0);
#endif
}

// ---------------------------------------------------------------------------
// Kernel 1: y = x @ W^T + b for W in {Wq,Wk,Wv} (selected by blockIdx.z).
// Workgroup = 256 threads (8 wave32), computes a 128x64 output tile.
// Each wave: 16 rows x 64 cols via 4 f32 WMMA accumulators; K in chunks of 32.
// WMMA mapping: D = A(16x4) x B(4x16) + C, A[m][k]=x[m][k], B[k][n]=W[n][k].
// B fragments come from an LDS-staged W tile (padded stride -> no conflicts).
// ---------------------------------------------------------------------------
__global__ __launch_bounds__(256) void aft_qkv_gemm(
    const float* __restrict__ X,
    const float* __restrict__ Wq, const float* __restrict__ bq,
    const float* __restrict__ Wk, const float* __restrict__ bk,
    const float* __restrict__ Wv, const float* __restrict__ bv,
    float* __restrict__ Q, float* __restrict__ K, float* __restrict__ V)
{
    const float* W; const float* bias; float* Y;
    if (blockIdx.z == 0)      { W = Wq; bias = bq; Y = Q; }
    else if (blockIdx.z == 1) { W = Wk; bias = bk; Y = K; }
    else                      { W = Wv; bias = bv; Y = V; }

    const int n0   = blockIdx.x * 64;    // output-column tile
    const int m0   = blockIdx.y * 128;   // output-row tile (flattened B*N)
    const int lane = threadIdx.x & 31;
    const int wave = threadIdx.x >> 5;
    const int mw   = m0 + wave * 16;     // this wave's 16 rows
    const int ln   = lane & 15;
    const int kb   = (lane >> 4) * 2;    // K sub-offset per ISA f32-A layout

    __shared__ float Wt[64][34];         // [n_local][k_local], +2 pad floats

    // accumulators preloaded with bias (broadcast over all 8 row-VGPRs)
    v8f acc[4];
    #pragma unroll
    for (int c = 0; c < 4; ++c) {
        float b0 = bias[n0 + c * 16 + ln];
        #pragma unroll
        for (int j = 0; j < 8; ++j) acc[c][j] = b0;
    }

    for (int kc = 0; kc < D_; kc += 32) {
        // cooperative stage: Wt[i][j] = W[n0+i][kc+j], 64x32 floats
        {
            const int i = threadIdx.x >> 2;        // 0..63
            const int j = (threadIdx.x & 3) * 8;   // 0,8,16,24
            const float* src = &W[(size_t)(n0 + i) * D_ + kc + j];
            v4f w0 = *(const v4f*)(src);
            v4f w1 = *(const v4f*)(src + 4);
            // 8B-granularity LDS stores (row stride 34 floats keeps 8B align)
            *(v2f*)&Wt[i][j + 0] = w0.lo;
            *(v2f*)&Wt[i][j + 2] = w0.hi;
            *(v2f*)&Wt[i][j + 4] = w1.lo;
            *(v2f*)&Wt[i][j + 6] = w1.hi;
        }
        __syncthreads();

        #pragma unroll
        for (int kk = 0; kk < 32; kk += 4) {
            // A fragment: contiguous float2 per lane (ISA 16x4 f32 layout)
            v2f a = *(const v2f*)&X[(size_t)(mw + ln) * D_ + kc + kk + kb];
            #pragma unroll
            for (int c = 0; c < 4; ++c) {
                v2f bf = *(const v2f*)&Wt[c * 16 + ln][kk + kb];
                acc[c] = __builtin_amdgcn_wmma_f32_16x16x4_f32(
                    false, a, false, bf, (short)0, acc[c], false, false);
            }
        }
        __syncthreads();
    }

    // epilogue: C/D layout -> m = (lane>>4)*8 + j, n = lane&15
    #pragma unroll
    for (int c = 0; c < 4; ++c) {
        #pragma unroll
        for (int j = 0; j < 8; ++j) {
            const int m = mw + (lane >> 4) * 8 + j;
            const int n = n0 + c * 16 + ln;
            Y[(size_t)m * D_ + n] = acc[c][j];
        }
    }
}

// ---------------------------------------------------------------------------
// Kernel 2: ew[t][s] = exp(pos_bias[t][s] - max_s pos_bias[t][s])
// One 256-thread block per row t.
// ---------------------------------------------------------------------------
__global__ __launch_bounds__(256) void aft_ew(
    const float* __restrict__ pb, float* __restrict__ ew)
{
    const int t = blockIdx.x;
    __shared__ float red[256];
    float m = -INFINITY;
    for (int s = threadIdx.x; s < N_; s += 256)
        m = fmaxf(m, pb[(size_t)t * N_ + s]);
    red[threadIdx.x] = m;
    __syncthreads();
    for (int off = 128; off > 0; off >>= 1) {
        if (threadIdx.x < off)
            red[threadIdx.x] = fmaxf(red[threadIdx.x], red[threadIdx.x + off]);
        __syncthreads();
    }
    const float rm = red[0];
    for (int s = threadIdx.x; s < N_; s += 256)
        ew[(size_t)t * N_ + s] = __expf(pb[(size_t)t * N_ + s] - rm);
}

// ---------------------------------------------------------------------------
// Kernel 3: kmax[b][d] = max_s k[b][s][d].  One thread per (b,d); loads
// coalesce across d.
// ---------------------------------------------------------------------------
__global__ __launch_bounds__(256) void aft_kmax(
    const float* __restrict__ K, float* __restrict__ kmax)
{
    const int tid = blockIdx.x * 256 + threadIdx.x;  // b*512 + d
    const int b = tid >> 9;
    const int d = tid & (D_ - 1);
    const float* p = K + (size_t)b * N_ * D_ + d;
    float m = -INFINITY;
    for (int s = 0; s < N_; ++s) m = fmaxf(m, p[(size_t)s * D_]);
    kmax[tid] = m;
}

// ---------------------------------------------------------------------------
// Kernel 4: in place: k <- exp(k - kmax[b][d]);  v <- ek * v
// ---------------------------------------------------------------------------
__global__ __launch_bounds__(256) void aft_ekv(
    float* __restrict__ K, float* __restrict__ V,
    const float* __restrict__ kmax)
{
    const size_t total = (size_t)B_ * N_ * D_;
    for (size_t i = (size_t)blockIdx.x * 256 + threadIdx.x; i < total;
         i += (size_t)gridDim.x * 256) {
        const int d = (int)(i & (D_ - 1));
        const int b = (int)(i / ((size_t)N_ * D_));
        const float e = __expf(K[i] - kmax[b * D_ + d]);
        K[i] = e;
        V[i] = e * V[i];
    }
}

// ---------------------------------------------------------------------------
// Kernel 5: fused attention: per batch b,
//   num = ew @ ekv, den = ew @ ek  (f32 WMMA, K=s over 1024)
//   out = sigmoid(q) * num / den   (fused epilogue; num/den never hit HBM)
// Workgroup = 256 threads, tile 128(t) x 64(d); each wave holds 4 num + 4 den
// accumulators sharing one A(ew) fragment per k-step (8 WMMAs per A load).
// ek/ekv tiles are streamed global->LDS by the Tensor Data Mover with a
// double-buffered TENSORcnt pipeline issued by wave 0.
// ---------------------------------------------------------------------------
__global__ __launch_bounds__(256) void aft_attn(
    const float* __restrict__ EW, const float* __restrict__ EK,
    const float* __restrict__ EKV, float* __restrict__ QO)
{
    const int d0   = blockIdx.x * 64;
    const int t0   = blockIdx.y * 128;
    const int b    = blockIdx.z;
    const int lane = threadIdx.x & 31;
    const int wave = threadIdx.x >> 5;
    const int tw   = t0 + wave * 16;
    const int ln   = lane & 15;
    const int kb   = (lane >> 4) * 2;

    // [stage][ek|ekv][s_local][d_local], 32KB total, TDM-filled
    __shared__ float tiles[2][2][32][64];
    const unsigned lds_base =
        (unsigned)(unsigned long long)(size_t)&tiles[0][0][0][0];
    const unsigned stage_bytes = 2u * 32u * 64u * 4u;   // 16KB per stage
    const unsigned tile_bytes  = 32u * 64u * 4u;        // 8KB per tensor

    v8f accN[4], accD[4];
    #pragma unroll
    for (int c = 0; c < 4; ++c)
        #pragma unroll
        for (int j = 0; j < 8; ++j) { accN[c][j] = 0.f; accD[c][j] = 0.f; }

    const size_t bbase = (size_t)b * N_ * D_;
    const bool issuer = (wave == 0);

    // prologue: TDM-issue chunk 0 into stage 0 (2 ops: ek, ekv)
    if (issuer) {
        tdm_load_tile_32x64_f32(lds_base + 0,
                                &EK[bbase + d0]);
        tdm_load_tile_32x64_f32(lds_base + tile_bytes,
                                &EKV[bbase + d0]);
    }

    for (int i = 0; i < N_ / 32; ++i) {
        const int cur = i & 1;
        const int sc  = i * 32;
        if (issuer) {
            if (i + 1 < N_ / 32) {
                // issue chunk i+1 into the other stage, then wait until the
                // current chunk's 2 TDM ops have landed (in-order completion)
                const unsigned sb = lds_base + (unsigned)(cur ^ 1) * stage_bytes;
                const size_t gn = bbase + (size_t)(sc + 32) * D_ + d0;
                tdm_load_tile_32x64_f32(sb + 0,          &EK[gn]);
                tdm_load_tile_32x64_f32(sb + tile_bytes, &EKV[gn]);
                __builtin_amdgcn_s_wait_tensorcnt(2);
            } else {
                __builtin_amdgcn_s_wait_tensorcnt(0);
            }
        }
        __syncthreads();   // current stage visible to all waves

        const float (*ekT)[64] = (const float (*)[64])tiles[cur][0];
        const float (*evT)[64] = (const float (*)[64])tiles[cur][1];

        #pragma unroll
        for (int kk = 0; kk < 32; kk += 4) {
            v2f a = *(const v2f*)&EW[(size_t)(tw + ln) * N_ + sc + kk + kb];
            #pragma unroll
            for (int c = 0; c < 4; ++c) {
                const int n = c * 16 + ln;
                v2f bn = { evT[kk + kb][n], evT[kk + kb + 1][n] };
                v2f bd = { ekT[kk + kb][n], ekT[kk + kb + 1][n] };
                accN[c] = __builtin_amdgcn_wmma_f32_16x16x4_f32(
                    false, a, false, bn, (short)0, accN[c], false, false);
                accD[c] = __builtin_amdgcn_wmma_f32_16x16x4_f32(
                    false, a, false, bd, (short)0, accD[c], false, false);
            }
        }
        __syncthreads();   // done reading this stage before TDM overwrites it
    }

    // fused epilogue: out = sigmoid(q) * num / den, overwriting q in d_out
    #pragma unroll
    for (int c = 0; c < 4; ++c) {
        #pragma unroll
        for (int j = 0; j < 8; ++j) {
            const int t = tw + (lane >> 4) * 8 + j;
            const int d = d0 + c * 16 + ln;
            const size_t idx = ((size_t)b * N_ + t) * D_ + d;
            const float qv  = QO[idx];
            const float sig = 1.0f / (1.0f + __expf(-qv));
            QO[idx] = sig * accN[c][j] / accD[c][j];
        }
    }
}

// ---------------------------------------------------------------------------
// Host launcher
// ---------------------------------------------------------------------------
extern "C" void kernel_launch(void* const* d_in, const int* in_sizes, int n_in,
                              void* d_out, int out_size, void* d_ws, size_t ws_size,
                              hipStream_t stream)
{
    (void)in_sizes; (void)n_in; (void)out_size; (void)ws_size;

    const float* x  = (const float*)d_in[0];
    const float* Wq = (const float*)d_in[1];
    const float* bq = (const float*)d_in[2];
    const float* Wk = (const float*)d_in[3];
    const float* bk = (const float*)d_in[4];
    const float* Wv = (const float*)d_in[5];
    const float* bv = (const float*)d_in[6];
    const float* pb = (const float*)d_in[7];

    float* q    = (float*)d_out;                 // q staged in d_out
    float* wsf  = (float*)d_ws;
    const size_t bnd = (size_t)B_ * N_ * D_;     // 16,777,216
    float* k    = wsf;                           // -> ek (in place)
    float* v    = wsf + bnd;                     // -> ekv (in place)
    float* ew   = wsf + 2 * bnd;                 // N*N
    float* kmax = ew + (size_t)N_ * N_;          // B*D

    // 1) QKV projections (z = 0:q, 1:k, 2:v)
    aft_qkv_gemm<<<dim3(D_ / 64, (B_ * N_) / 128, 3), 256, 0, stream>>>(
        x, Wq, bq, Wk, bk, Wv, bv, q, k, v);

    // 2) ew = exp(pos_bias - rowmax)
    aft_ew<<<dim3(N_), 256, 0, stream>>>(pb, ew);

    // 3) kmax[b][d]
    aft_kmax<<<dim3((B_ * D_) / 256), 256, 0, stream>>>(k, kmax);

    // 4) k <- ek, v <- ekv (in place)
    aft_ekv<<<dim3(16384), 256, 0, stream>>>(k, v, kmax);

    // 5) fused num/den GEMMs + sigmoid(q)*num/den epilogue (TDM pipeline)
    aft_attn<<<dim3(D_ / 64, N_ / 128, B_), 256, 0, stream>>>(ew, k, v, q);
}